// BERTgridGenerator_49074296324476
// MI455X (gfx1250) — compile-verified
//
#include <hip/hip_runtime.h>

typedef __attribute__((ext_vector_type(2))) float v2f;
typedef __attribute__((ext_vector_type(4))) float v4f;
typedef __attribute__((ext_vector_type(8))) float v8f;

#define B_    8
#define S_    512
#define D_    768
#define NSEG  128
#define GH_   128
#define GW_   128

// ---------------------------------------------------------------------------
// Kernel 1: per-(batch,segment) inverse counts.  invcnt[b*NSEG+n] = 1/max(c,1)
// (exact fp32 division done once per segment here, so the WMMA kernel's
//  epilogue is a pure multiply -- no v_div_scale/v_rcp chains there)
// ---------------------------------------------------------------------------
__global__ void bg_count_kernel(const long long* __restrict__ seg,
                                const int* __restrict__ mask,
                                float* __restrict__ invcnt) {
    int b = blockIdx.x;                  // B_ blocks, 128 threads
    __shared__ int ssg[S_];
    __shared__ int smk[S_];
    for (int i = threadIdx.x; i < S_; i += blockDim.x) {
        ssg[i] = (int)seg[(size_t)b * S_ + i];
        smk[i] = mask[(size_t)b * S_ + i];
    }
    __syncthreads();
    int n = threadIdx.x;
    int c = 0;
    for (int s = 0; s < S_; ++s) c += (ssg[s] == n) ? smk[s] : 0;
    invcnt[b * NSEG + n] = 1.0f / fmaxf((float)c, 1.0f);
}

// ---------------------------------------------------------------------------
// Kernel 2: per-pixel owner = max box index covering (h,w), -1 if none.
// ---------------------------------------------------------------------------
__global__ void bg_owner_kernel(const int* __restrict__ coor,
                                int* __restrict__ owner) {
    int b = blockIdx.x;                  // B_ blocks, 256 threads
    __shared__ int4 box[NSEG];
    for (int i = threadIdx.x; i < NSEG; i += blockDim.x)
        box[i] = ((const int4*)coor)[(size_t)b * NSEG + i];
    __syncthreads();
    for (int p = threadIdx.x; p < GH_ * GW_; p += blockDim.x) {
        int h = p / GW_, w = p % GW_;
        int own = -1;
        #pragma unroll 4
        for (int n = 0; n < NSEG; ++n) {
            int4 bb = box[n];           // x0,y0,x1,y1
            if (w >= bb.x && h >= bb.y && w < bb.z && h < bb.w) own = n;  // ascending => max
        }
        owner[((size_t)b * GH_ + h) * GW_ + w] = own;
    }
}

// ---------------------------------------------------------------------------
// Kernel 3: seg_emb = (W @ E) * invcnt via V_WMMA_F32_16X16X4_F32.
// One wave computes all 128 segments x one 16-wide d-tile.
// A (16x4 f32): lanes 0-15 hold M=0..15 with VGPR0->K=0, VGPR1->K=1;
//               lanes 16-31 -> K=2,3.   (built on the fly from seg/mask)
// B (4x16 f32): VGPR j, lane l: K = j + 2*(l>=16), N = l&15.
// C (16x16 f32): VGPR r, lanes 0-15 -> M=r, lanes 16-31 -> M=r+8, N = l&15.
// ---------------------------------------------------------------------------
__global__ void bg_segemb_wmma_kernel(const float* __restrict__ emb,
                                      const int* __restrict__ mask,
                                      const long long* __restrict__ seg,
                                      const float* __restrict__ invcnt,
                                      float* __restrict__ seg_emb) {
    // grid: B_ * 6 blocks of 256 threads (8 waves); each wave owns one d-tile
    int b    = blockIdx.x / 6;
    int wave = threadIdx.x >> 5;
    int dt   = (blockIdx.x % 6) * 8 + wave;   // 0..47
    int d0   = dt * 16;
    int lane = threadIdx.x & 31;
    int half = lane >> 4;                     // 0: lanes 0-15, 1: lanes 16-31
    int row  = lane & 15;

    v8f acc[8] = {};                          // 8 seg-tiles x (16x16 f32 tile)

    const float*     Eb = emb  + (size_t)b * S_ * D_;
    const long long* Sb = seg  + (size_t)b * S_;
    const int*       Mb = mask + (size_t)b * S_;

    for (int k0 = 0; k0 < S_; k0 += 4) {
        int t0 = k0 + half * 2;               // this half-wave's K pair
        int s0 = (int)Sb[t0];
        int s1 = (int)Sb[t0 + 1];
        float w0 = (float)Mb[t0];
        float w1 = (float)Mb[t0 + 1];

        v2f bm;
        bm.x = Eb[(size_t)(t0 + 0) * D_ + d0 + row];   // VGPR0: K = 2*half
        bm.y = Eb[(size_t)(t0 + 1) * D_ + d0 + row];   // VGPR1: K = 2*half+1

        #pragma unroll
        for (int m = 0; m < 8; ++m) {
            int base = m * 16 + row;          // segment id of this A row
            v2f am;
            am.x = (s0 == base) ? w0 : 0.0f;
            am.y = (s1 == base) ? w1 : 0.0f;
            acc[m] = __builtin_amdgcn_wmma_f32_16x16x4_f32(
                false, am, false, bm, (short)0, acc[m], false, false);
        }
    }

    const float* Ib = invcnt + (size_t)b * NSEG;
    float*       Ob = seg_emb + (size_t)b * NSEG * D_;
    #pragma unroll
    for (int m = 0; m < 8; ++m) {
        #pragma unroll
        for (int r = 0; r < 8; ++r) {
            int segid = m * 16 + half * 8 + r;
            Ob[(size_t)segid * D_ + d0 + row] = acc[m][r] * Ib[segid];
        }
    }
}

// ---------------------------------------------------------------------------
// Kernel 4: grid[b,d,h,w] = owner>=0 ? seg_emb[b,owner,d] : 0.
// One block per (b,h); owner row in LDS; per-wave contiguous 512B stores.
// The 402 MB output exceeds the 192 MB L2, so stores are NON-TEMPORAL
// (native clang vector type so __builtin_nontemporal_store accepts it):
// keeps L2 resident for the seg_emb gather table (3 MB) instead of letting
// the write stream thrash it.  This kernel bounds runtime (~17us @ 23.3 TB/s).
// ---------------------------------------------------------------------------
__global__ void bg_grid_fill_kernel(const float* __restrict__ seg_emb,
                                    const int* __restrict__ owner,
                                    float* __restrict__ grid) {
    int b = blockIdx.x / GH_;
    int h = blockIdx.x % GH_;
    __shared__ int sown[GW_];
    if (threadIdx.x < GW_)
        sown[threadIdx.x] = owner[((size_t)b * GH_ + h) * GW_ + threadIdx.x];
    __syncthreads();

    int w4    = threadIdx.x & 31;    // 32 float4 spans the 128-wide row
    int dlane = threadIdx.x >> 5;    // 0..7 (one wave per d-phase)
    int o0 = sown[w4 * 4 + 0];
    int o1 = sown[w4 * 4 + 1];
    int o2 = sown[w4 * 4 + 2];
    int o3 = sown[w4 * 4 + 3];

    const float* Sb = seg_emb + (size_t)b * NSEG * D_;
    float*       Gb = grid + (size_t)b * D_ * GH_ * GW_ + (size_t)h * GW_;

    for (int d = dlane; d < D_; d += 8) {
        v4f v;
        v.x = (o0 >= 0) ? Sb[(size_t)o0 * D_ + d] : 0.0f;
        v.y = (o1 >= 0) ? Sb[(size_t)o1 * D_ + d] : 0.0f;
        v.z = (o2 >= 0) ? Sb[(size_t)o2 * D_ + d] : 0.0f;
        v.w = (o3 >= 0) ? Sb[(size_t)o3 * D_ + d] : 0.0f;
        __builtin_nontemporal_store(v, (v4f*)(Gb + (size_t)d * GH_ * GW_ + w4 * 4));
    }
}

// ---------------------------------------------------------------------------
extern "C" void kernel_launch(void* const* d_in, const int* in_sizes, int n_in,
                              void* d_out, int out_size, void* d_ws, size_t ws_size,
                              hipStream_t stream) {
    const float*     emb  = (const float*)d_in[0];      // [B,S,D] f32
    const int*       mask = (const int*)d_in[1];        // [B,S] i32
    const long long* seg  = (const long long*)d_in[2];  // [B,S] i64
    const int*       coor = (const int*)d_in[3];        // [B,N,4] i32
    // d_in[4..6]: num_segments / grid_h / grid_w scalars (compile-time here)

    float* seg_emb = (float*)d_out;                               // [B,N,D]
    float* grid    = (float*)d_out + (size_t)B_ * NSEG * D_;      // [B,D,H,W]

    float* invcnt = (float*)d_ws;                                 // 4 KB
    int*   owner  = (int*)((char*)d_ws + B_ * NSEG * sizeof(float)); // 512 KB

    bg_count_kernel<<<B_, 128, 0, stream>>>(seg, mask, invcnt);
    bg_owner_kernel<<<B_, 256, 0, stream>>>(coor, owner);
    bg_segemb_wmma_kernel<<<B_ * 6, 256, 0, stream>>>(emb, mask, seg, invcnt, seg_emb);
    bg_grid_fill_kernel<<<B_ * GH_, 256, 0, stream>>>(seg_emb, owner, grid);
}